// MPC_25829933318848
// MI455X (gfx1250) — compile-verified
//
#include <hip/hip_runtime.h>
#include <cmath>

// ---- problem constants (must match reference) ----
constexpr int   NX    = 64;
constexpr int   NU    = 16;
constexpr int   NH    = 64;    // horizon N
constexpr int   BATCH = 256;   // B
constexpr int   ITERS = 2;
constexpr float DT    = 0.05f;

typedef __attribute__((ext_vector_type(2))) float v2f;
typedef __attribute__((ext_vector_type(8))) float v8f;

// ---------------------------------------------------------------------------
// WMMA micro-GEMM: acc(16x16) += opA(A) * opB(B), K steps of V_WMMA_F32_16X16X4_F32
// A fragment layout (ISA 7.12.2, 32-bit A 16x4): lanes 0-15 hold M=0..15,
//   vgpr0 = K(+0|+2), vgpr1 = K(+1|+3) per lane-half.
// B fragment mirrored (KxN, col per lane, lane-half selects K pair).
// C/D: 8 VGPRs, lanes 0-15 -> M=r, lanes 16-31 -> M=8+r, N=lane&15.
// TA: A stored transposed (K-major, element (m,k) at A[k*lda+m])
// TB: B stored transposed (N-major, element (k,n) at B[n*ldb+k])
// ---------------------------------------------------------------------------
template<bool TA, bool TB, int K>
__device__ inline v8f wmma_acc(const float* __restrict__ Ab, int lda,
                               const float* __restrict__ Bb, int ldb, v8f acc) {
  const int lane = threadIdx.x & 31;
  const int idx  = lane & 15;
  const int h2   = (lane >> 4) << 1;   // 0 or 2
#pragma unroll
  for (int k = 0; k < K; k += 4) {
    v2f a, b;
    if (TA) {
      a[0] = Ab[(k + h2 + 0) * lda + idx];
      a[1] = Ab[(k + h2 + 1) * lda + idx];
    } else {
      a[0] = Ab[idx * lda + k + h2 + 0];
      a[1] = Ab[idx * lda + k + h2 + 1];
    }
    if (TB) {
      b[0] = Bb[idx * ldb + k + h2 + 0];
      b[1] = Bb[idx * ldb + k + h2 + 1];
    } else {
      b[0] = Bb[(k + h2 + 0) * ldb + idx];
      b[1] = Bb[(k + h2 + 1) * ldb + idx];
    }
    acc = __builtin_amdgcn_wmma_f32_16x16x4_f32(false, a, false, b,
                                                (short)0, acc, false, false);
  }
  return acc;
}

__device__ inline v8f load_ctile(const float* __restrict__ C, int ldc) {
  const int lane = threadIdx.x & 31;
  const int n  = lane & 15;
  const int mo = (lane >> 4) * 8;
  v8f c;
#pragma unroll
  for (int r = 0; r < 8; ++r) c[r] = C[(mo + r) * ldc + n];
  return c;
}

__device__ inline void store_ctile(float* __restrict__ C, int ldc, v8f c) {
  const int lane = threadIdx.x & 31;
  const int n  = lane & 15;
  const int mo = (lane >> 4) * 8;
#pragma unroll
  for (int r = 0; r < 8; ++r) C[(mo + r) * ldc + n] = c[r];
}

// ---------------------------------------------------------------------------
// KKT kernel: for all N*B rows compute z = xs@A^T + u@Bm^T, t = tanh(z),
// s2, xres, rx, ru. Fx/Fu are NOT materialized (Fx = I + DT*diag(s2)*A).
// 128 rows per workgroup (8 waves x 16-row M-tile), all B operands in LDS.
// ---------------------------------------------------------------------------
__global__ __launch_bounds__(256) void kkt_kernel(
    const float* __restrict__ x, const float* __restrict__ u,
    const float* __restrict__ lmd,
    const float* __restrict__ A,  const float* __restrict__ Bm,
    const float* __restrict__ Qx, const float* __restrict__ R,
    const float* __restrict__ xref,
    float* __restrict__ s2g, float* __restrict__ xresg,
    float* __restrict__ rxg, float* __restrict__ rug)
{
  constexpr int ROWS = 128;
  __shared__ float sA[NX * NX], sBm[NX * NU], sQx[NX * NX], sR[NU * NU];
  __shared__ float sXref[NX];
  __shared__ float sX[ROWS * NX];   // xs, later xs - xref
  __shared__ float sU[ROWS * NU];
  __shared__ float sW[ROWS * NX];   // z -> w = DT*s2*lmd1

  const int tid  = threadIdx.x;
  const int w    = tid >> 5;
  const int lane = tid & 31;
  const int r0   = blockIdx.x * ROWS;      // global row = n*BATCH + b

  for (int i = tid; i < NX * NX; i += 256) { sA[i] = A[i]; sQx[i] = Qx[i]; }
  for (int i = tid; i < NX * NU; i += 256) sBm[i] = Bm[i];
  for (int i = tid; i < NU * NU; i += 256) sR[i]  = R[i];
  if (tid < NX) sXref[tid] = xref[tid];
  for (int i = tid; i < ROWS * NX; i += 256) sX[i] = x[(size_t)r0 * NX + i];
  for (int i = tid; i < ROWS * NU; i += 256) sU[i] = u[(size_t)r0 * NU + i];
  __syncthreads();

  // z = xs @ A^T + u @ Bm^T   (per-wave 16x64 output, 4 N-tiles)
#pragma unroll
  for (int nt = 0; nt < 4; ++nt) {
    v8f acc = {};
    acc = wmma_acc<false, true, NX>(&sX[w * 16 * NX], NX, &sA[nt * 16 * NX], NX, acc);
    acc = wmma_acc<false, true, NU>(&sU[w * 16 * NU], NU, &sBm[nt * 16 * NU], NU, acc);
    store_ctile(&sW[w * 16 * NX + nt * 16], NX, acc);
  }
  __syncthreads();

  // elementwise: t, s2, xres, w = DT*s2*lmd[n+1], xs -= xref
  for (int i = tid; i < ROWS * NX; i += 256) {
    const int row = i >> 6, col = i & 63;
    const size_t r = (size_t)(r0 + row);
    const float z   = sW[i];
    const float t   = tanhf(z);
    const float s2v = 1.0f - t * t;
    const float xsv = sX[i];
    xresg[r * NX + col] = xsv + DT * t - x[(r + BATCH) * NX + col];
    s2g[r * NX + col]   = s2v;
    sW[i] = DT * s2v * lmd[(r + BATCH) * NX + col];
    sX[i] = xsv - sXref[col];
  }
  __syncthreads();

  // rx = (xs-xref)@Qx + w@A + lmd[n+1] - lmd[n]
#pragma unroll
  for (int nt = 0; nt < 4; ++nt) {
    v8f acc = {};
    acc = wmma_acc<false, false, NX>(&sX[w * 16 * NX], NX, &sQx[nt * 16], NX, acc);
    acc = wmma_acc<false, false, NX>(&sW[w * 16 * NX], NX, &sA[nt * 16],  NX, acc);
    const int n = lane & 15, mo = (lane >> 4) * 8;
#pragma unroll
    for (int rr = 0; rr < 8; ++rr) {
      const size_t r = (size_t)(r0 + w * 16 + mo + rr);
      const int col  = nt * 16 + n;
      rxg[r * NX + col] = acc[rr] + lmd[(r + BATCH) * NX + col] - lmd[r * NX + col];
    }
  }
  // ru = u@R + w@Bm
  {
    v8f acc = {};
    acc = wmma_acc<false, false, NU>(&sU[w * 16 * NU], NU, sR,  NU, acc);
    acc = wmma_acc<false, false, NX>(&sW[w * 16 * NX], NX, sBm, NU, acc);
    const int n = lane & 15, mo = (lane >> 4) * 8;
#pragma unroll
    for (int rr = 0; rr < 8; ++rr) {
      const size_t r = (size_t)(r0 + w * 16 + mo + rr);
      rug[r * NU + n] = acc[rr];
    }
  }
}

// rN = (x[N]-xref)@Vf - lmd[N]
__global__ __launch_bounds__(256) void rn_kernel(
    const float* __restrict__ x, const float* __restrict__ lmd,
    const float* __restrict__ Vf, const float* __restrict__ xref,
    float* __restrict__ rNg)
{
  constexpr int ROWS = 128;
  __shared__ float sVf[NX * NX], sXref[NX], sX[ROWS * NX];
  const int tid = threadIdx.x, w = tid >> 5, lane = tid & 31;
  const int r0 = blockIdx.x * ROWS;
  const size_t base = ((size_t)NH * BATCH + r0) * NX;

  for (int i = tid; i < NX * NX; i += 256) sVf[i] = Vf[i];
  if (tid < NX) sXref[tid] = xref[tid];
  __syncthreads();
  for (int i = tid; i < ROWS * NX; i += 256) sX[i] = x[base + i] - sXref[i & 63];
  __syncthreads();

#pragma unroll
  for (int nt = 0; nt < 4; ++nt) {
    v8f acc = {};
    acc = wmma_acc<false, false, NX>(&sX[w * 16 * NX], NX, &sVf[nt * 16], NX, acc);
    const int n = lane & 15, mo = (lane >> 4) * 8;
#pragma unroll
    for (int rr = 0; rr < 8; ++rr) {
      const int row = w * 16 + mo + rr;
      const int col = nt * 16 + n;
      rNg[(size_t)(r0 + row) * NX + col] = acc[rr] - lmd[base + (size_t)row * NX + col];
    }
  }
}

// ---------------------------------------------------------------------------
// Backward Riccati: one workgroup (8 waves) per batch element, LDS-resident P.
// Per step: PFx=P@Fx, PFu=P@Fu, Qxx=Qx+Fx^T PFx (overwrites P), Quu=R+Fu^T PFu,
// Qxu=Fx^T PFu, 16x16 Gauss-Jordan solve with 65 RHS, Pn=Qxx+Qxu@K, symmetrize.
// Stores K,k for the forward pass.
// ---------------------------------------------------------------------------
__global__ __launch_bounds__(256) void bwd_kernel(
    const float* __restrict__ A,  const float* __restrict__ Bm,
    const float* __restrict__ Qx, const float* __restrict__ R,
    const float* __restrict__ Vf,
    const float* __restrict__ s2g, const float* __restrict__ xresg,
    const float* __restrict__ rxg, const float* __restrict__ rug,
    const float* __restrict__ rNg,
    float* __restrict__ Kg, float* __restrict__ kg)
{
  constexpr int LDAUG = 81;   // [Quu(16) | Qxu^T(64) | lu(1)]
  __shared__ float sA[NX * NX], sBm[NX * NU], sQx[NX * NX], sR[NU * NU];
  __shared__ float sP[NX * NX], sFx[NX * NX], sFu[NX * NU];
  __shared__ float sPFx[NX * NX], sPFu[NX * NU], sQxu[NX * NU], sK[NU * NX];
  __shared__ float sAug[NU * LDAUG];
  __shared__ float sS2[NX], sXres[NX], sV[NX], sS[NX], sLx[NX];
  __shared__ float sFac[NU], skv[NU];

  const int b = blockIdx.x;
  const int tid = threadIdx.x;
  const int w = tid >> 5;

  for (int i = tid; i < NX * NX; i += 256) { sA[i] = A[i]; sQx[i] = Qx[i]; sP[i] = Vf[i]; }
  for (int i = tid; i < NX * NU; i += 256) sBm[i] = Bm[i];
  for (int i = tid; i < NU * NU; i += 256) sR[i] = R[i];
  if (tid < NX) sS[tid] = rNg[(size_t)b * NX + tid];
  __syncthreads();

  for (int n = NH - 1; n >= 0; --n) {
    const size_t rbase = (size_t)(n * BATCH + b);
    if (tid < NX) {
      sS2[tid]   = s2g[rbase * NX + tid];
      sXres[tid] = xresg[rbase * NX + tid];
      if (n > 0) {
        __builtin_prefetch(&s2g[(rbase - BATCH) * NX + tid], 0, 1);
        __builtin_prefetch(&rxg[(rbase - BATCH) * NX + tid], 0, 1);
      }
    }
    __syncthreads();

    // Fx = I + DT*diag(s2)*A ; Fu = DT*diag(s2)*Bm
    for (int i = tid; i < NX * NX; i += 256) {
      const int ri = i >> 6, ci = i & 63;
      sFx[i] = DT * sS2[ri] * sA[i] + (ri == ci ? 1.0f : 0.0f);
    }
    for (int i = tid; i < NX * NU; i += 256) sFu[i] = DT * sS2[i >> 4] * sBm[i];
    // v = P@xres + s
    if (tid < NX) {
      float acc = sS[tid];
      for (int j = 0; j < NX; ++j) acc += sP[tid * NX + j] * sXres[j];
      sV[tid] = acc;
    }
    __syncthreads();

    // PFx (16 tiles) + PFu (4 tiles), wave-uniform tile split
    for (int t = w; t < 20; t += 8) {
      if (t < 16) {
        const int mt = t >> 2, nt = t & 3;
        v8f acc = {};
        acc = wmma_acc<false, false, NX>(&sP[mt * 16 * NX], NX, &sFx[nt * 16], NX, acc);
        store_ctile(&sPFx[mt * 16 * NX + nt * 16], NX, acc);
      } else {
        const int mt = t - 16;
        v8f acc = {};
        acc = wmma_acc<false, false, NX>(&sP[mt * 16 * NX], NX, sFu, NU, acc);
        store_ctile(&sPFu[mt * 16 * NU], NU, acc);
      }
    }
    __syncthreads();

    // lx = rx + Fx^T v ; lu = ru + Fu^T v (into Aug col 80)
    if (tid < NX) {
      float acc = rxg[rbase * NX + tid];
      for (int j = 0; j < NX; ++j) acc += sFx[j * NX + tid] * sV[j];
      sLx[tid] = acc;
    } else if (tid < NX + NU) {
      const int i = tid - NX;
      float acc = rug[rbase * NU + i];
      for (int j = 0; j < NX; ++j) acc += sFu[j * NU + i] * sV[j];
      sAug[i * LDAUG + 80] = acc;
    }
    // Qxx (16 tiles, overwrites P) + Quu (1) + Qxu (4)
    for (int t = w; t < 21; t += 8) {
      if (t < 16) {
        const int mt = t >> 2, nt = t & 3;
        v8f acc = load_ctile(&sQx[mt * 16 * NX + nt * 16], NX);
        acc = wmma_acc<true, false, NX>(&sFx[mt * 16], NX, &sPFx[nt * 16], NX, acc);
        store_ctile(&sP[mt * 16 * NX + nt * 16], NX, acc);
      } else if (t == 16) {
        v8f acc = load_ctile(sR, NU);
        acc = wmma_acc<true, false, NX>(sFu, NU, sPFu, NU, acc);
        store_ctile(sAug, LDAUG, acc);
      } else {
        const int mt = t - 17;
        v8f acc = {};
        acc = wmma_acc<true, false, NX>(&sFx[mt * 16], NX, sPFu, NU, acc);
        store_ctile(&sQxu[mt * 16 * NU], NU, acc);
      }
    }
    __syncthreads();

    // RHS: Qxu^T into Aug cols 16..79
    for (int i = tid; i < NU * NX; i += 256) {
      const int ri = i >> 6, ci = i & 63;
      sAug[ri * LDAUG + 16 + ci] = sQxu[ci * NU + ri];
    }
    __syncthreads();

    // Gauss-Jordan on [Quu | RHS]  (Quu SPD, no pivoting needed)
    for (int j = 0; j < NU; ++j) {
      if (tid < NU) sFac[tid] = sAug[tid * LDAUG + j];
      __syncthreads();
      const float invp = 1.0f / sFac[j];
      for (int c = tid; c < LDAUG; c += 256) sAug[j * LDAUG + c] *= invp;
      __syncthreads();
      for (int e = tid; e < NU * LDAUG; e += 256) {
        const int ri = e / LDAUG, c = e - ri * LDAUG;
        if (ri != j) sAug[ri * LDAUG + c] -= sFac[ri] * sAug[j * LDAUG + c];
      }
      __syncthreads();
    }

    // K = -sol[:, :64], k = -sol[:, 64]
    for (int i = tid; i < NU * NX; i += 256) {
      const int ri = i >> 6, ci = i & 63;
      const float kv = -sAug[ri * LDAUG + 16 + ci];
      sK[ri * NX + ci] = kv;
      Kg[rbase * NU * NX + i] = kv;
    }
    if (tid < NU) {
      const float kvv = -sAug[tid * LDAUG + 80];
      skv[tid] = kvv;
      kg[rbase * NU + tid] = kvv;
    }
    __syncthreads();

    // Pn = Qxx + Qxu@K  (in place on sP, K-dim 16)
    for (int t = w; t < 16; t += 8) {
      const int mt = t >> 2, nt = t & 3;
      v8f acc = load_ctile(&sP[mt * 16 * NX + nt * 16], NX);
      acc = wmma_acc<false, false, NU>(&sQxu[mt * 16 * NU], NU, &sK[nt * 16], NX, acc);
      store_ctile(&sP[mt * 16 * NX + nt * 16], NX, acc);
    }
    __syncthreads();

    // symmetrize ; sn = lx + Qxu@k
    for (int p = tid; p < NX * NX; p += 256) {
      const int ri = p >> 6, ci = p & 63;
      if (ri < ci) {
        const float av = 0.5f * (sP[ri * NX + ci] + sP[ci * NX + ri]);
        sP[ri * NX + ci] = av;
        sP[ci * NX + ri] = av;
      }
    }
    if (tid < NX) {
      float acc = sLx[tid];
      for (int j = 0; j < NU; ++j) acc += sQxu[tid * NU + j] * skv[j];
      sS[tid] = acc;
    }
    __syncthreads();
  }
}

// ---------------------------------------------------------------------------
// Forward rollout: du = K dx + k ; dx' = dx + DT*s2*(A dx + Bm du) + xres.
// Updates x,u in place, stores dx for the costate sweep.
// ---------------------------------------------------------------------------
__global__ __launch_bounds__(64) void fwd_kernel(
    const float* __restrict__ x0,
    const float* __restrict__ A, const float* __restrict__ Bm,
    const float* __restrict__ s2g, const float* __restrict__ xresg,
    const float* __restrict__ Kg, const float* __restrict__ kg,
    float* __restrict__ x, float* __restrict__ u, float* __restrict__ dxg)
{
  __shared__ float sA[NX * NX], sBm[NX * NU], sdx[NX], sdu[NU];
  const int b = blockIdx.x, tid = threadIdx.x;

  for (int i = tid; i < NX * NX; i += 64) sA[i] = A[i];
  for (int i = tid; i < NX * NU; i += 64) sBm[i] = Bm[i];
  sdx[tid] = x0[(size_t)b * NX + tid] - x[(size_t)b * NX + tid];  // dx0 = -(x[0]-x0)
  __syncthreads();

  for (int n = 0; n < NH; ++n) {
    const size_t rbase = (size_t)(n * BATCH + b);
    const float dxi = sdx[tid];
    x[rbase * NX + tid] += dxi;
    dxg[rbase * NX + tid] = dxi;
    if (tid < NU) {
      float acc = kg[rbase * NU + tid];
      const float* Kr = &Kg[rbase * NU * NX + (size_t)tid * NX];
      for (int j = 0; j < NX; ++j) acc += Kr[j] * sdx[j];
      sdu[tid] = acc;
    }
    float adx = 0.0f;
    for (int j = 0; j < NX; ++j) adx += sA[tid * NX + j] * sdx[j];
    __syncthreads();
    float bmdu = 0.0f;
    for (int j = 0; j < NU; ++j) bmdu += sBm[tid * NU + j] * sdu[j];
    if (tid < NU) u[rbase * NU + tid] += sdu[tid];
    const float dxn = dxi + DT * s2g[rbase * NX + tid] * (adx + bmdu)
                          + xresg[rbase * NX + tid];
    __syncthreads();
    sdx[tid] = dxn;
    __syncthreads();
  }
  const size_t rN = (size_t)(NH * BATCH + b);
  x[rN * NX + tid] += sdx[tid];
  dxg[rN * NX + tid] = sdx[tid];
}

// ---------------------------------------------------------------------------
// Costate sweep (replaces storing all P_n):
//   dlmd_N = Vf dx_N + rN ;  dlmd_n = Qx dx_n + rx_n + Fx^T dlmd_{n+1}
//   with Fx^T g = g + DT * A^T (s2 .* g).   lmd += dlmd.
// ---------------------------------------------------------------------------
__global__ __launch_bounds__(64) void lmd_kernel(
    const float* __restrict__ A, const float* __restrict__ Qx,
    const float* __restrict__ Vf,
    const float* __restrict__ s2g, const float* __restrict__ rxg,
    const float* __restrict__ rNg, const float* __restrict__ dxg,
    float* __restrict__ lmd)
{
  __shared__ float sA[NX * NX], sQx[NX * NX], sdx[NX], sw[NX];
  const int b = blockIdx.x, tid = threadIdx.x;

  for (int i = tid; i < NX * NX; i += 64) { sA[i] = A[i]; sQx[i] = Qx[i]; }
  const size_t rN = (size_t)(NH * BATCH + b);
  sdx[tid] = dxg[rN * NX + tid];
  __syncthreads();

  float g = rNg[(size_t)b * NX + tid];
  for (int j = 0; j < NX; ++j) g += Vf[tid * NX + j] * sdx[j];
  lmd[rN * NX + tid] += g;

  for (int n = NH - 1; n >= 0; --n) {
    const size_t rbase = (size_t)(n * BATCH + b);
    __syncthreads();
    sw[tid]  = s2g[rbase * NX + tid] * g;
    sdx[tid] = dxg[rbase * NX + tid];
    __syncthreads();
    float gn = rxg[rbase * NX + tid] + g;
    for (int j = 0; j < NX; ++j)
      gn += sQx[tid * NX + j] * sdx[j] + DT * sA[j * NX + tid] * sw[j];
    g = gn;
    lmd[rbase * NX + tid] += g;
  }
}

// ---------------------------------------------------------------------------
extern "C" void kernel_launch(void* const* d_in, const int* in_sizes, int n_in,
                              void* d_out, int out_size, void* d_ws, size_t ws_size,
                              hipStream_t stream) {
  (void)in_sizes; (void)n_in; (void)out_size; (void)ws_size;
  const float* x0_i  = (const float*)d_in[0];
  const float* x_i   = (const float*)d_in[1];
  const float* u_i   = (const float*)d_in[2];
  const float* lmd_i = (const float*)d_in[3];
  const float* A     = (const float*)d_in[4];
  const float* Bm    = (const float*)d_in[5];
  const float* Qx    = (const float*)d_in[6];
  const float* R     = (const float*)d_in[7];
  const float* Vf    = (const float*)d_in[8];
  const float* xref  = (const float*)d_in[9];

  const size_t nX   = (size_t)(NH + 1) * BATCH * NX;
  const size_t nU   = (size_t)NH * BATCH * NU;
  float* xo = (float*)d_out;
  float* uo = xo + nX;
  float* lo = uo + nU;

  float* p = (float*)d_ws;
  float* s2g   = p; p += (size_t)NH * BATCH * NX;
  float* xresg = p; p += (size_t)NH * BATCH * NX;
  float* rxg   = p; p += (size_t)NH * BATCH * NX;
  float* rug   = p; p += (size_t)NH * BATCH * NU;
  float* rNg   = p; p += (size_t)BATCH * NX;
  float* Kg    = p; p += (size_t)NH * BATCH * NU * NX;
  float* kg    = p; p += (size_t)NH * BATCH * NU;
  float* dxg   = p; p += nX;

  hipMemcpyAsync(xo, x_i,   nX * sizeof(float), hipMemcpyDeviceToDevice, stream);
  hipMemcpyAsync(uo, u_i,   nU * sizeof(float), hipMemcpyDeviceToDevice, stream);
  hipMemcpyAsync(lo, lmd_i, nX * sizeof(float), hipMemcpyDeviceToDevice, stream);

  for (int it = 0; it < ITERS; ++it) {
    kkt_kernel<<<(NH * BATCH) / 128, 256, 0, stream>>>(
        xo, uo, lo, A, Bm, Qx, R, xref, s2g, xresg, rxg, rug);
    rn_kernel<<<BATCH / 128, 256, 0, stream>>>(xo, lo, Vf, xref, rNg);
    bwd_kernel<<<BATCH, 256, 0, stream>>>(
        A, Bm, Qx, R, Vf, s2g, xresg, rxg, rug, rNg, Kg, kg);
    fwd_kernel<<<BATCH, 64, 0, stream>>>(
        x0_i, A, Bm, s2g, xresg, Kg, kg, xo, uo, dxg);
    lmd_kernel<<<BATCH, 64, 0, stream>>>(
        A, Qx, Vf, s2g, rxg, rNg, dxg, lo);
  }
}